// EKVConv2d_27144193310980
// MI455X (gfx1250) — compile-verified
//
#include <hip/hip_runtime.h>

// EKV conv: out[b,oc,p] = ALPHA*TIA*Σ_k [ sp((vg-θ)inv)² − sp((vg-θ−VD)inv)² ]
// Transcendental-VALU-bound. All math in base-2 domain (native v_exp_f32 /
// v_log_f32, no ln fixup chains); ln2² and ALPHA*TIA folded into the WMMA B.
// WMMA f32 16x16x4 acts as co-executing K-reducer/scaler into the 16x16 C tile.
// Theta staged into LDS with CDNA5 async global->LDS b128 copies.

typedef float v2f __attribute__((ext_vector_type(2)));
typedef float v8f __attribute__((ext_vector_type(8)));
typedef int   v4i __attribute__((ext_vector_type(4)));

#define INVQ2   36.99218053561445f    // (1/(N_FACTOR*VT)) * log2(e)
#define VDI2    3.699218053561445f    // VD * INVQ2
#define CLIP2   43.2808512266689f     // 30 * log2(e)
#define OSCALE2 1351.2741015f         // ALPHA*TIA_GAIN * ln(2)^2

#if defined(__has_builtin)
#if __has_builtin(__builtin_amdgcn_global_load_async_to_lds_b128) && \
    __has_builtin(__builtin_amdgcn_s_wait_asynccnt)
#define USE_ASYNC_LDS 1
#endif
#if __has_builtin(__builtin_amdgcn_exp2f)
#define EXP2F(x) __builtin_amdgcn_exp2f(x)
#endif
#endif
#ifndef EXP2F
#define EXP2F(x) exp2f(x)
#endif

__device__ __forceinline__ float sp2diff2(float vgi2, float th) {
    // base-2 domain: z = (vg - th)*INVQ*log2e
    float z1  = fmaf(th, -INVQ2, vgi2);
    float z1c = fminf(fmaxf(z1, -CLIP2), CLIP2);          // v_med3_num_f32
    float z2c = fminf(fmaxf(z1 - VDI2, -CLIP2), CLIP2);
    float l1  = __log2f(1.0f + EXP2F(z1c));               // native exp2/log2
    float l2  = __log2f(1.0f + EXP2F(z2c));
    return fmaf(l1, l1, -(l2 * l2));                      // * ln2^2 folded in B
}

__global__ __launch_bounds__(128)
void ekv_conv2d_wmma(const float* __restrict__ x,
                     const float* __restrict__ theta,
                     float* __restrict__ out) {
    __shared__ alignas(16) float th_lds[64 * 72];   // theta K-chunk [oc][72]
    __shared__ float xg_lds[32 * 3 * 18];           // prescaled x halo [c][row][col]

    const int tid  = threadIdx.x;
    const int bid  = blockIdx.x;
    const int b    = bid >> 6;               // 8 batches
    const int t    = bid & 63;               // 64 strips per image
    const int h    = t >> 1;                 // row 0..31
    const int w0   = (t & 1) << 4;           // col 0 or 16

    // ---- x halo tile -> LDS, prescaled by INVQ*log2e, zero-padded ----
    for (int i = tid; i < 32 * 3 * 18; i += 128) {
        int c   = i / 54;
        int rem = i - c * 54;
        int row = rem / 18;
        int col = rem - row * 18;
        int gr  = h - 1 + row;
        int gc  = w0 - 1 + col;
        float v = 0.0f;
        if ((unsigned)gr < 32u && (unsigned)gc < 32u)
            v = x[((b * 32 + c) * 32 + gr) * 32 + gc] * INVQ2;
        xg_lds[i] = v;
    }

    const int lane  = tid & 31;
    const int wave  = tid >> 5;
    const int m     = lane & 15;             // pixel-in-strip AND wmma column id
    const int khalf = lane >> 4;             // K pair selector of 16x4 A tile
    const int oc0   = wave << 4;             // 4 waves cover 64 channels

    // Precompute column-select B values (loop invariant -> hoisted to VGPRs).
    float bsel[16];
    #pragma unroll
    for (int n = 0; n < 16; ++n)
        bsel[n] = (m == n) ? OSCALE2 : 0.0f;

    v8f acc = {};                            // 16x16 f32 C/D tile (8 VGPRs)

    const float4* th4 = (const float4*)theta;

    for (int s = 0; s < 4; ++s) {            // theta staged in 4 K-chunks of 72
        __syncthreads();
        float4* thl4 = (float4*)th_lds;
        for (int i = tid; i < 64 * 18; i += 128) {
            int oc = i / 18;
            int j4 = i - oc * 18;
#ifdef USE_ASYNC_LDS
            __builtin_amdgcn_global_load_async_to_lds_b128(
                (__attribute__((address_space(1))) v4i*)&th4[oc * 72 + s * 18 + j4],
                (__attribute__((address_space(3))) v4i*)&thl4[oc * 18 + j4],
                0, 0);
#else
            thl4[oc * 18 + j4] = th4[oc * 72 + s * 18 + j4];
#endif
        }
#ifdef USE_ASYNC_LDS
        __builtin_amdgcn_s_wait_asynccnt(0);
#endif
        __syncthreads();

        for (int kc = s * 18; kc < s * 18 + 18; ++kc) {
            const int kb = kc * 4 + khalf * 2;   // this lane's first K index
            int k0 = kb,     c0 = k0 / 9, r0 = k0 - c0 * 9, i0 = r0 / 3, j0 = r0 - i0 * 3;
            int k1 = kb + 1, c1 = k1 / 9, r1 = k1 - c1 * 9, i1 = r1 / 3, j1 = r1 - i1 * 3;
            const float vg0 = xg_lds[(c0 * 3 + i0) * 18 + m + j0];
            const float vg1 = xg_lds[(c1 * 3 + i1) * 18 + m + j1];

            const int kloc = kb - s * 72;        // offset within staged chunk
            #pragma unroll
            for (int n = 0; n < 16; ++n) {
                const float2 th = *(const float2*)&th_lds[(oc0 + n) * 72 + kloc];
                v2f a;
                a.x = sp2diff2(vg0, th.x);
                a.y = sp2diff2(vg1, th.y);
                v2f bb; bb.x = bsel[n]; bb.y = bsel[n];
                // D[m][n'] += Σ_k A[m][k] * B[k][n']  — accumulates column n
                acc = __builtin_amdgcn_wmma_f32_16x16x4_f32(
                          false, a, false, bb, (short)0, acc, false, false);
            }
        }
    }

    // ---- store: acc VGPR r holds (M = r + 8*khalf, N = m) of the 16x16 tile
    const int oc = oc0 + m;
    float* op = out + ((size_t)(b * 64 + oc) << 10) + h * 32 + w0 + khalf * 8;
    #pragma unroll
    for (int r = 0; r < 8; ++r)
        op[r] = acc[r];
}

extern "C" void kernel_launch(void* const* d_in, const int* in_sizes, int n_in,
                              void* d_out, int out_size, void* d_ws, size_t ws_size,
                              hipStream_t stream) {
    const float* x     = (const float*)d_in[0];   // (8,32,32,32) f32
    const float* theta = (const float*)d_in[1];   // (64,288)     f32
    float*       out   = (float*)d_out;           // (8,64,32,32) f32
    (void)in_sizes; (void)n_in; (void)out_size; (void)d_ws; (void)ws_size;
    ekv_conv2d_wmma<<<dim3(512), dim3(128), 0, stream>>>(x, theta, out);
}